// SyntheticSampler_4552665334265
// MI455X (gfx1250) — compile-verified
//
#include <hip/hip_runtime.h>
#include <stdint.h>

// Problem constants from the reference
#define D_ITEM   64     // floats per item row (256 bytes = 16 float4 chunks)
#define MAX_LEN  100
#define D_CTX    16

#define BLOCK    256
#define CHUNKS_PER_THREAD 8          // 8 outstanding async b128 ops per lane
#define CHUNKS_PER_ROW   (D_ITEM/4) // 16 float4 chunks per item row

// ---------------------------------------------------------------------------
// Kernel 0: build a 256-byte zero page in workspace (padded slots gather this)
// ---------------------------------------------------------------------------
__global__ void zero_page_kernel(float4* __restrict__ ws) {
    int t = threadIdx.x;
    if (t < CHUNKS_PER_ROW) ws[t] = make_float4(0.f, 0.f, 0.f, 0.f);
}

// ---------------------------------------------------------------------------
// Kernel 1: items gather via CDNA5 async global->LDS->global streaming.
// Each thread moves 8 float4 chunks. Chunk c maps to slot = c/16 (one
// (i,j) output row) and part = c%16 (which float4 of the 64-float row).
// Valid slots gather from the universe; padded slots gather the zero page,
// so EXEC stays all-ones through the async pipeline.
// ---------------------------------------------------------------------------
__global__ __launch_bounds__(BLOCK)
void items_gather_async(const float* __restrict__ universe,
                        const int*   __restrict__ idx,
                        const int*   __restrict__ sizes,
                        const float* __restrict__ zero_page,
                        float*       __restrict__ out_items,
                        int total_chunks)            // n*MAX_LEN*16 < 2^31
{
    __shared__ float4 lds_buf[BLOCK * CHUNKS_PER_THREAD];   // 32 KB / block

    const int block_base = blockIdx.x * (BLOCK * CHUNKS_PER_THREAD);
    const unsigned lds_base = (unsigned)(uintptr_t)(&lds_buf[0]); // LDS offset

    // ---- Phase 1: issue async gathers (global -> LDS), 8 in flight -------
    #pragma unroll
    for (int k = 0; k < CHUNKS_PER_THREAD; ++k) {
        int c = block_base + k * BLOCK + (int)threadIdx.x;
        if (c >= total_chunks) break;
        int slot = c >> 4;                 // output (i,j) row id
        int part = c & (CHUNKS_PER_ROW - 1);
        int i = slot / MAX_LEN;
        int j = slot - i * MAX_LEN;
        bool valid = j < sizes[i];
        const float* src = valid
            ? (universe + (size_t)idx[slot] * D_ITEM + (size_t)(part * 4))
            : (zero_page + (size_t)(part * 4));
        unsigned lds_addr = lds_base + (unsigned)(k * BLOCK + (int)threadIdx.x) * 16u;
        asm volatile("global_load_async_to_lds_b128 %0, %1, off"
                     :: "v"(lds_addr), "v"(src)
                     : "memory");
    }

    // All async loads landed in LDS before we read it back
    asm volatile("s_wait_asynccnt 0" ::: "memory");

    // ---- Phase 2: stream LDS -> output, non-temporal (write-once stream,
    // keeps the 2.56 MB universe resident in the 192 MB L2) ----------------
    #pragma unroll
    for (int k = 0; k < CHUNKS_PER_THREAD; ++k) {
        int c = block_base + k * BLOCK + (int)threadIdx.x;
        if (c >= total_chunks) break;
        unsigned lds_addr = lds_base + (unsigned)(k * BLOCK + (int)threadIdx.x) * 16u;
        float* dst = out_items + (size_t)c * 4u;
        asm volatile("global_store_async_from_lds_b128 %0, %1, off th:TH_STORE_NT"
                     :: "v"(dst), "v"(lds_addr)
                     : "memory");
    }
    asm volatile("s_wait_asynccnt 0" ::: "memory");
}

// ---------------------------------------------------------------------------
// Kernel 2: validity mask (N, MAX_LEN) -> 1.0 / 0.0
// ---------------------------------------------------------------------------
__global__ __launch_bounds__(BLOCK)
void mask_kernel(const int* __restrict__ sizes,
                 float* __restrict__ out_mask,
                 int total)                 // n * MAX_LEN
{
    int t = blockIdx.x * BLOCK + (int)threadIdx.x;
    if (t < total) {
        int i = t / MAX_LEN;
        int j = t - i * MAX_LEN;
        out_mask[t] = (j < sizes[i]) ? 1.0f : 0.0f;
    }
}

// ---------------------------------------------------------------------------
// Kernel 3: context clip to [-3, 3], float4 vectorized
// ---------------------------------------------------------------------------
__global__ __launch_bounds__(BLOCK)
void ctx_kernel(const float4* __restrict__ in,
                float4* __restrict__ out,
                int total4)                // n * D_CTX / 4
{
    int t = blockIdx.x * BLOCK + (int)threadIdx.x;
    if (t < total4) {
        float4 v = in[t];
        v.x = fminf(fmaxf(v.x, -3.0f), 3.0f);
        v.y = fminf(fmaxf(v.y, -3.0f), 3.0f);
        v.z = fminf(fmaxf(v.z, -3.0f), 3.0f);
        v.w = fminf(fmaxf(v.w, -3.0f), 3.0f);
        out[t] = v;
    }
}

// ---------------------------------------------------------------------------
extern "C" void kernel_launch(void* const* d_in, const int* in_sizes, int n_in,
                              void* d_out, int out_size, void* d_ws, size_t ws_size,
                              hipStream_t stream) {
    const float* universe = (const float*)d_in[0];   // (POOL, 64) fp32
    const float* ctx_in   = (const float*)d_in[1];   // (N, 16)   fp32
    const int*   idx      = (const int*)  d_in[2];   // (N, 100)  i32
    const int*   sizes    = (const int*)  d_in[3];   // (N,)      i32

    const int n_rows = in_sizes[3];                  // N

    float* out = (float*)d_out;
    float* out_items = out;                                        // N*100*64
    size_t items_elems = (size_t)n_rows * MAX_LEN * D_ITEM;
    float* out_mask = out + items_elems;                           // N*100
    float* out_ctx  = out_mask + (size_t)n_rows * MAX_LEN;         // N*16

    float* zero_page = (float*)d_ws;                 // 256 bytes of scratch

    // 0) zero page for padded slots (re-created every launch: deterministic)
    zero_page_kernel<<<1, 64, 0, stream>>>((float4*)zero_page);

    // 1) items gather (bulk of the 1.68 GB of traffic)
    int total_chunks = n_rows * MAX_LEN * CHUNKS_PER_ROW;          // < 2^31
    int gblocks = (total_chunks + BLOCK * CHUNKS_PER_THREAD - 1) /
                  (BLOCK * CHUNKS_PER_THREAD);
    items_gather_async<<<gblocks, BLOCK, 0, stream>>>(
        universe, idx, sizes, zero_page, out_items, total_chunks);

    // 2) mask
    int total_mask = n_rows * MAX_LEN;
    mask_kernel<<<(total_mask + BLOCK - 1) / BLOCK, BLOCK, 0, stream>>>(
        sizes, out_mask, total_mask);

    // 3) clipped context
    int total4 = (n_rows * D_CTX) / 4;
    ctx_kernel<<<(total4 + BLOCK - 1) / BLOCK, BLOCK, 0, stream>>>(
        (const float4*)ctx_in, (float4*)out_ctx, total4);
}